// ERGCNLayer_33526514713105
// MI455X (gfx1250) — compile-verified
//
#include <hip/hip_runtime.h>
#include <stdint.h>

#define DVEC 64     // feature dim (fixed by reference)
#define RMAX 200    // relation count (fixed by reference); LDS sized for this

typedef float v4f __attribute__((ext_vector_type(4)));   // native vector -> NT builtins ok

// ---------------------------------------------------------------------------
// Kernel: per-edge message + scatter-add.  (Defined first so the disasm
// snippet shows this kernel's instruction stream.)
//   - weight/attention tables (2 x R x 64 f32 = 102.4 KB) staged in LDS via
//     CDNA5 async global->LDS copies (ASYNCcnt), exploiting the 320KB WGP LDS.
//   - 16 lanes per edge, one float4 per lane (global_load_b128 / ds_load_b128).
//   - single-use streams (e, src, dst, rel) loaded non-temporally so the
//     gathered h rows and the atomically-updated out rows stay L2-resident.
//   - agent-scope relaxed f32 atomics -> hardware global_atomic_add_f32.
// ---------------------------------------------------------------------------
__global__ __launch_bounds__(256)
void ergcn_edge_kernel(const float* __restrict__ h,
                       const float* __restrict__ e,
                       const float* __restrict__ weight,
                       const float* __restrict__ attention,
                       const int*   __restrict__ src,
                       const int*   __restrict__ dst,
                       const int*   __restrict__ rel,
                       float*       __restrict__ out,
                       int E, int R)
{
    __shared__ float s_w[RMAX * DVEC];   // 51,200 B
    __shared__ float s_a[RMAX * DVEC];   // 51,200 B  (total 102,400 B < 320KB WGP LDS)

    const int tid  = threadIdx.x;
    const int nvec = (R * DVEC) / 4;     // float4 chunks per table (3200 for R=200)

    // --- async stage both tables into LDS (gfx1250 async-tensor path) ---
    for (int i = tid; i < nvec; i += (int)blockDim.x) {
        unsigned lds_w = (unsigned)(uintptr_t)(&s_w[4 * i]);   // AS(3) byte offset
        unsigned lds_a = (unsigned)(uintptr_t)(&s_a[4 * i]);
        unsigned goff  = (unsigned)(16u * (unsigned)i);        // byte offset from base
        asm volatile("global_load_async_to_lds_b128 %0, %1, %2"
                     :: "v"(lds_w), "v"(goff), "s"(weight) : "memory");
        asm volatile("global_load_async_to_lds_b128 %0, %1, %2"
                     :: "v"(lds_a), "v"(goff), "s"(attention) : "memory");
    }
    asm volatile("s_wait_asynccnt 0" ::: "memory");   // my wave's async copies done
    __syncthreads();                                  // all waves' copies visible

    // --- edge loop: 16 lanes per edge, float4 per lane ---
    const int group = tid >> 4;          // 0..15 : edge slot within block
    const int lane  = tid & 15;          // 0..15 : float4 slot within feature row
    const int col   = lane * 4;          // starting float within the 64-wide row

    const int estart  = (int)blockIdx.x * 16 + group;
    const int estride = (int)gridDim.x * 16;

    for (int k = estart; k < E; k += estride) {
        // single-use index streams: non-temporal
        const int s = __builtin_nontemporal_load(src + k);
        const int d = __builtin_nontemporal_load(dst + k);
        const int r = __builtin_nontemporal_load(rel + k);

        // h rows are reused ~16x across edges -> keep cacheable (L2-resident)
        const v4f hv = *reinterpret_cast<const v4f*>(h + (size_t)s * DVEC + col);
        // e rows are touched exactly once -> non-temporal
        const v4f ev = __builtin_nontemporal_load(
            reinterpret_cast<const v4f*>(e + (size_t)k * DVEC + col));
        const v4f wv = *reinterpret_cast<const v4f*>(&s_w[r * DVEC + col]);
        const v4f av = *reinterpret_cast<const v4f*>(&s_a[r * DVEC + col]);

        v4f m;
        m.x = fmaf(hv.x, wv.x, ev.x * av.x);
        m.y = fmaf(hv.y, wv.y, ev.y * av.y);
        m.z = fmaf(hv.z, wv.z, ev.z * av.z);
        m.w = fmaf(hv.w, wv.w, ev.w * av.w);

        float* o = out + (size_t)d * DVEC + col;
        __hip_atomic_fetch_add(o + 0, m.x, __ATOMIC_RELAXED, __HIP_MEMORY_SCOPE_AGENT);
        __hip_atomic_fetch_add(o + 1, m.y, __ATOMIC_RELAXED, __HIP_MEMORY_SCOPE_AGENT);
        __hip_atomic_fetch_add(o + 2, m.z, __ATOMIC_RELAXED, __HIP_MEMORY_SCOPE_AGENT);
        __hip_atomic_fetch_add(o + 3, m.w, __ATOMIC_RELAXED, __HIP_MEMORY_SCOPE_AGENT);
    }
}

// ---------------------------------------------------------------------------
// Kernel: out = h   (residual init, float4-vectorized streaming copy)
// ---------------------------------------------------------------------------
__global__ __launch_bounds__(256)
void ergcn_init_kernel(const float* __restrict__ h, float* __restrict__ out, int n4) {
    int i = blockIdx.x * blockDim.x + threadIdx.x;
    if (i < n4) {
        reinterpret_cast<v4f*>(out)[i] = reinterpret_cast<const v4f*>(h)[i];
    }
}

// ---------------------------------------------------------------------------
extern "C" void kernel_launch(void* const* d_in, const int* in_sizes, int n_in,
                              void* d_out, int out_size, void* d_ws, size_t ws_size,
                              hipStream_t stream) {
    (void)n_in; (void)d_ws; (void)ws_size; (void)out_size;

    const float* h         = (const float*)d_in[0];   // [N, 64]
    const float* e         = (const float*)d_in[1];   // [E, 64]
    const float* weight    = (const float*)d_in[2];   // [R, 64]
    const float* attention = (const float*)d_in[3];   // [R, 64]
    const int*   src       = (const int*)d_in[4];     // [E]
    const int*   dst       = (const int*)d_in[5];     // [E]
    const int*   rel       = (const int*)d_in[6];     // [E]
    float*       out       = (float*)d_out;           // [N, 64]

    const int ND = in_sizes[0];        // N * 64
    const int E  = in_sizes[4];
    const int R  = in_sizes[2] / DVEC; // 200

    // out = h
    const int n4 = ND / 4;
    ergcn_init_kernel<<<(n4 + 255) / 256, 256, 0, stream>>>(h, out, n4);

    // scatter-add messages; bounded grid so LDS table staging amortizes
    int blocks = (E + 15) / 16;
    if (blocks > 2048) blocks = 2048;
    if (blocks < 1)    blocks = 1;
    ergcn_edge_kernel<<<blocks, 256, 0, stream>>>(h, e, weight, attention,
                                                  src, dst, rel, out, E, R);
}